// ShapeMatchingConstraint_25821343383615
// MI455X (gfx1250) — compile-verified
//
#include <hip/hip_runtime.h>

typedef float v2f __attribute__((ext_vector_type(2)));
typedef float v8f __attribute__((ext_vector_type(8)));

#define WAVES_PB   8
#define THREADS    256
#define NPTS       64
#define ELEMS      192          // 64*3 floats per batch
#define STEPS      48           // 192 / K(=4) WMMA steps
#define DEF_STRIDE 196          // padded per-batch LDS stride (floats): 196%64=4 -> conflict-free
#define RV_STRIDE  20           // per-batch R/v stride (floats), 16B aligned, bank-friendly

// LDS layout (float offsets)
#define B_OFF     0                          // 48*64 = 3072 floats: WMMA B fragments per (step,lane)
#define REST_OFF  3072                       // 64 x float4 (rest positions, padded)
#define DEF_OFF   3328                       // 8 waves * 16 batches * 196
#define RV_OFF    (3328 + WAVES_PB*16*DEF_STRIDE)   // 28416: D-tile exchange then R/v, 320 fl/wave
#define LDS_FLOATS (RV_OFF + WAVES_PB*16*RV_STRIDE) // 30976
#define LDS_BYTES  (LDS_FLOATS * 4)                 // 123904 B (2 blocks fit in 320KB WGP LDS)

// One cyclic-Jacobi rotation on symmetric H (p,q), third index r; V <- V*J.
#define JROT(hpp,hqq,hpq,hrp,hrq, vp0,vp1,vp2, vq0,vq1,vq2)                         \
  {                                                                                  \
    float apq = hpq;                                                                 \
    if (fabsf(apq) > 1e-30f) {                                                       \
      float tau = (hqq - hpp) * 0.5f / apq;                                          \
      float tj  = copysignf(1.0f, tau) / (fabsf(tau) + sqrtf(fmaf(tau, tau, 1.0f))); \
      float cj  = 1.0f / sqrtf(fmaf(tj, tj, 1.0f));                                  \
      float sj  = tj * cj;                                                           \
      hpp = hpp - tj * apq;                                                          \
      hqq = hqq + tj * apq;                                                          \
      hpq = 0.0f;                                                                    \
      float u1 = hrp, u2 = hrq;                                                      \
      hrp = cj*u1 - sj*u2;  hrq = sj*u1 + cj*u2;                                     \
      u1 = vp0; u2 = vq0; vp0 = cj*u1 - sj*u2; vq0 = sj*u1 + cj*u2;                  \
      u1 = vp1; u2 = vq1; vp1 = cj*u1 - sj*u2; vq1 = sj*u1 + cj*u2;                  \
      u1 = vp2; u2 = vq2; vp2 = cj*u1 - sj*u2; vq2 = sj*u1 + cj*u2;                  \
    }                                                                                \
  }

__global__ __launch_bounds__(THREADS)
void shape_match_wmma_kernel(const float* __restrict__ rest,
                             const float* __restrict__ masses,
                             const float* __restrict__ defp,
                             float* __restrict__ outp, int nBatch)
{
  extern __shared__ float lds[];
  const int tid  = threadIdx.x;
  const int wave = tid >> 5;
  const int lane = tid & 31;

  // ---------- setup: rest COM / total mass (uniform -> scalar loads, redundant per thread)
  float msum = 0.f, rcx = 0.f, rcy = 0.f, rcz = 0.f;
  for (int n = 0; n < NPTS; ++n) {
    float m = masses[n];
    msum += m;
    rcx = fmaf(m, rest[3*n+0], rcx);
    rcy = fmaf(m, rest[3*n+1], rcy);
    rcz = fmaf(m, rest[3*n+2], rcz);
  }
  const float invM = 1.0f / msum;
  rcx *= invM; rcy *= invM; rcz *= invM;

  // rest positions, float4-padded, for the output phase
  for (int n = tid; n < NPTS; n += THREADS) {
    lds[REST_OFF + 4*n + 0] = rest[3*n+0];
    lds[REST_OFF + 4*n + 1] = rest[3*n+1];
    lds[REST_OFF + 4*n + 2] = rest[3*n+2];
    lds[REST_OFF + 4*n + 3] = 0.f;
  }

  // Build WMMA B-operand fragments.  K-dim flattens (n,i): k = 3n+i.
  // Columns: c = 3*i'+j (c<9): delta(i,i') * m_n*(rest[n][j]-rcom[j])  -> A_pq
  //          c = 9+i'  (c<12): delta(i,i') * m_n/Msum                  -> def COM
  // Fragment layout per ISA: VGPR0 = {K0 | K2}, VGPR1 = {K1 | K3} across lane halves.
  // Stored per (step s, lane l, vgpr j) at idx = s*64 + 2*l + j.
  for (int idx = tid; idx < STEPS*64; idx += THREADS) {
    int s = idx >> 6, rem = idx & 63, l = rem >> 1, j = rem & 1;
    int K = 4*s + ((l >> 4) << 1) + j;
    int n = K / 3, i = K - 3*n;
    int col = l & 15;
    float val = 0.f;
    if (col < 9) {
      int ip = col / 3, jj = col - 3*ip;
      if (i == ip) {
        float rc = (jj == 0) ? rcx : ((jj == 1) ? rcy : rcz);
        val = masses[n] * (rest[3*n+jj] - rc);
      }
    } else if (col < 12) {
      if (i == col - 9) val = masses[n] * invM;
    }
    lds[B_OFF + idx] = val;
  }
  __syncthreads();

  const int batchB = (blockIdx.x * WAVES_PB + wave) * 16;
  const bool valid = (batchB + 15) < nBatch || batchB < nBatch;
  const int  myb   = lane & 15;
  const int  half2 = (lane >> 4) << 1;     // A-fragment K sub-offset: lanes16-31 hold K+2,K+3

  // ---------- phase 1: 48 chained f32 WMMAs; D[m, 0..11] = [A_pq(9) | def_com(3)] for 16 batches
  v8f acc0 = {0.f,0.f,0.f,0.f,0.f,0.f,0.f,0.f};
  v8f acc1 = {0.f,0.f,0.f,0.f,0.f,0.f,0.f,0.f};
  if (valid) {
    const float* gdef = defp + (size_t)(batchB + myb) * ELEMS + half2;
    float*       dl   = &lds[DEF_OFF + wave*(16*DEF_STRIDE) + myb*DEF_STRIDE + half2];
    const float* bl   = &lds[B_OFF + 2*lane];
    for (int s = 0; s < STEPS; s += 2) {
      v2f a0 = *(const v2f*)(gdef + 4*s);
      v2f a1 = *(const v2f*)(gdef + 4*s + 4);
      v2f b0 = *(const v2f*)(bl + 64*s);
      v2f b1 = *(const v2f*)(bl + 64*s + 64);
      *(v2f*)(dl + 4*s)     = a0;              // stage def tile in LDS (avoid 2nd HBM read)
      *(v2f*)(dl + 4*s + 4) = a1;
      acc0 = __builtin_amdgcn_wmma_f32_16x16x4_f32(false, a0, false, b0, (short)0, acc0, false, false);
      acc1 = __builtin_amdgcn_wmma_f32_16x16x4_f32(false, a1, false, b1, (short)0, acc1, false, false);
    }
  }
  v8f D = acc0 + acc1;

  // D tile -> LDS exchange ([batch*16 + col]); VGPR r holds M=r (lanes 0-15) / M=r+8 (16-31)
  if (valid) {
    float* dx = &lds[RV_OFF + wave*(16*RV_STRIDE)];
    const int mh = (lane >> 4) << 3;
#pragma unroll
    for (int r = 0; r < 8; ++r) dx[(r + mh)*16 + myb] = D[r];
  }
  __syncthreads();

  // ---------- phase 2: per-batch polar decomposition (lane k<16 owns batch k)
  if (valid && lane < 16) {
    const float* dx = &lds[RV_OFF + wave*(16*RV_STRIDE) + lane*16];
    float4 m0 = *(const float4*)(dx);
    float4 m1 = *(const float4*)(dx + 4);
    float4 m2 = *(const float4*)(dx + 8);
    float a0=m0.x, a1=m0.y, a2=m0.z, a3=m0.w, a4=m1.x, a5=m1.y, a6=m1.z, a7=m1.w, a8=m2.x;
    float cmx=m2.y, cmy=m2.z, cmz=m2.w;

    // H = A^T A  (A[k][i] = a[3k+i])
    float h00 = a0*a0 + a3*a3 + a6*a6;
    float h01 = a0*a1 + a3*a4 + a6*a7;
    float h02 = a0*a2 + a3*a5 + a6*a8;
    float h11 = a1*a1 + a4*a4 + a7*a7;
    float h12 = a1*a2 + a4*a5 + a7*a8;
    float h22 = a2*a2 + a5*a5 + a8*a8;
    float v00=1.f,v01=0.f,v02=0.f, v10=0.f,v11=1.f,v12=0.f, v20=0.f,v21=0.f,v22=1.f;
#pragma unroll
    for (int sw = 0; sw < 5; ++sw) {
      JROT(h00,h11,h01, h02,h12, v00,v10,v20, v01,v11,v21);
      JROT(h00,h22,h02, h01,h12, v00,v10,v20, v02,v12,v22);
      JROT(h11,h22,h12, h01,h02, v01,v11,v21, v02,v12,v22);
    }
    float l0=h00, l1=h11, l2=h22;
    float c0 = 1.0f / fmaxf(sqrtf(fmaxf(l0,0.f)), 1e-20f);
    float c1 = 1.0f / fmaxf(sqrtf(fmaxf(l1,0.f)), 1e-20f);
    float c2 = 1.0f / fmaxf(sqrtf(fmaxf(l2,0.f)), 1e-20f);
    float detA = a0*(a4*a8 - a5*a7) - a1*(a3*a8 - a5*a6) + a2*(a3*a7 - a4*a6);
    float sg = (detA < 0.f) ? -1.f : 1.f;           // flip applied to smallest sigma
    if (l0 <= l1 && l0 <= l2)      c0 *= sg;
    else if (l1 <= l2)             c1 *= sg;
    else                           c2 *= sg;
    // Kmat = V diag(c) V^T (symmetric); column signs/order of V are irrelevant here
    float k00 = c0*v00*v00 + c1*v01*v01 + c2*v02*v02;
    float k01 = c0*v00*v10 + c1*v01*v11 + c2*v02*v12;
    float k02 = c0*v00*v20 + c1*v01*v21 + c2*v02*v22;
    float k11 = c0*v10*v10 + c1*v11*v11 + c2*v12*v12;
    float k12 = c0*v10*v20 + c1*v11*v21 + c2*v12*v22;
    float k22 = c0*v20*v20 + c1*v21*v21 + c2*v22*v22;
    // R = A * Kmat
    float R00 = a0*k00 + a1*k01 + a2*k02;
    float R01 = a0*k01 + a1*k11 + a2*k12;
    float R02 = a0*k02 + a1*k12 + a2*k22;
    float R10 = a3*k00 + a4*k01 + a5*k02;
    float R11 = a3*k01 + a4*k11 + a5*k12;
    float R12 = a3*k02 + a4*k12 + a5*k22;
    float R20 = a6*k00 + a7*k01 + a8*k02;
    float R21 = a6*k01 + a7*k11 + a8*k12;
    float R22 = a6*k02 + a7*k12 + a8*k22;
    // goal_pre = R*rest_n + (rcom + def_com - 2*R*rcom); fold STIFFNESS=0.9 in
    float Rrx = R00*rcx + R01*rcy + R02*rcz;
    float Rry = R10*rcx + R11*rcy + R12*rcz;
    float Rrz = R20*rcx + R21*rcy + R22*rcz;
    float ux = rcx + cmx - 2.f*Rrx;
    float uy = rcy + cmy - 2.f*Rry;
    float uz = rcz + cmz - 2.f*Rrz;
    float* rv = &lds[RV_OFF + wave*(16*RV_STRIDE) + lane*RV_STRIDE];
    rv[0]=0.9f*R00; rv[1]=0.9f*R01; rv[2]=0.9f*R02;
    rv[3]=0.9f*R10; rv[4]=0.9f*R11; rv[5]=0.9f*R12;
    rv[6]=0.9f*R20; rv[7]=0.9f*R21; rv[8]=0.9f*R22;
    rv[9]=0.9f*ux;  rv[10]=0.9f*uy; rv[11]=0.9f*uz;
  }
  __syncthreads();

  // ---------- phase 3: stream goal out; lane owns batch (lane&15), 96-elem half (lane>>4)
  if (valid) {
    const int h = lane >> 4;
    const float* rvp = &lds[RV_OFF + wave*(16*RV_STRIDE) + myb*RV_STRIDE];
    float4 q0 = *(const float4*)(rvp);
    float4 q1 = *(const float4*)(rvp + 4);
    float4 q2 = *(const float4*)(rvp + 8);
    float R00=q0.x,R01=q0.y,R02=q0.z,R10=q0.w,R11=q1.x,R12=q1.y,R20=q1.z,R21=q1.w,R22=q2.x;
    float vx=q2.y, vy=q2.z, vz=q2.w;
    const float*  dl = &lds[DEF_OFF + wave*(16*DEF_STRIDE) + myb*DEF_STRIDE + h*96];
    float*        og = outp + (size_t)(batchB + myb) * ELEMS + h*96;
    const float4* r4 = (const float4*)&lds[REST_OFF];
    const int nb = h << 5;
#pragma unroll 4
    for (int t = 0; t < 16; ++t) {            // 6 elements (= 2 points) per iter, i is static
      float4 ra = r4[nb + 2*t];
      float4 rb = r4[nb + 2*t + 1];
      int e = 6*t;
      float2 d0 = *(const float2*)(dl + e);
      float2 d1 = *(const float2*)(dl + e + 2);
      float2 d2 = *(const float2*)(dl + e + 4);
      float g0 = fmaf(R00, ra.x, fmaf(R01, ra.y, fmaf(R02, ra.z, vx)));
      float g1 = fmaf(R10, ra.x, fmaf(R11, ra.y, fmaf(R12, ra.z, vy)));
      float g2 = fmaf(R20, ra.x, fmaf(R21, ra.y, fmaf(R22, ra.z, vz)));
      float g3 = fmaf(R00, rb.x, fmaf(R01, rb.y, fmaf(R02, rb.z, vx)));
      float g4 = fmaf(R10, rb.x, fmaf(R11, rb.y, fmaf(R12, rb.z, vy)));
      float g5 = fmaf(R20, rb.x, fmaf(R21, rb.y, fmaf(R22, rb.z, vz)));
      float2 o0; o0.x = fmaf(0.1f, d0.x, g0); o0.y = fmaf(0.1f, d0.y, g1);
      float2 o1; o1.x = fmaf(0.1f, d1.x, g2); o1.y = fmaf(0.1f, d1.y, g3);
      float2 o2; o2.x = fmaf(0.1f, d2.x, g4); o2.y = fmaf(0.1f, d2.y, g5);
      *(float2*)(og + e)     = o0;
      *(float2*)(og + e + 2) = o1;
      *(float2*)(og + e + 4) = o2;
    }
  }
}

extern "C" void kernel_launch(void* const* d_in, const int* in_sizes, int n_in,
                              void* d_out, int out_size, void* d_ws, size_t ws_size,
                              hipStream_t stream) {
  const float* rest    = (const float*)d_in[0];   // (64,3)
  const float* masses  = (const float*)d_in[1];   // (64,)
  const float* defrmed = (const float*)d_in[2];   // (B,64,3)
  float* outp = (float*)d_out;
  int nBatch = in_sizes[2] / ELEMS;               // 65536
  int tiles  = (nBatch + 15) / 16;
  int blocks = (tiles + WAVES_PB - 1) / WAVES_PB; // 512
  shape_match_wmma_kernel<<<blocks, THREADS, LDS_BYTES, stream>>>(
      rest, masses, defrmed, outp, nBatch);
}